// NLFNet_2628519985779
// MI455X (gfx1250) — compile-verified
//
#include <hip/hip_runtime.h>

// ---------------- problem constants ----------------
#define NB    16
#define CH    3
#define HH    512
#define WW    512
#define RAD   4                 // window 9 -> radius 4
#define BAND  16                // output rows per workgroup
#define TROWS (BAND + 2*RAD)    // 24 halo rows
#define BLK1  512               // 16 waves of 32
#define BLK2  256
#define NPIX  (NB*HH*WW)        // 4,194,304
#define GRID2 2048

typedef __attribute__((ext_vector_type(2))) float v2f;
typedef __attribute__((ext_vector_type(8))) float v8f;

#ifdef __has_builtin
#if __has_builtin(__builtin_amdgcn_wmma_f32_16x16x4_f32)
#define HAVE_WMMA_F32 1
#endif
#endif
#ifndef HAVE_WMMA_F32
#define HAVE_WMMA_F32 0
#endif

__device__ __forceinline__ float fieldval(int f, float a, float b, float c) {
  if (f == 0) return a;
  if (f == 1) return b;
  if (f == 2) return c;
  if (f == 3) return a*a + b*b + c*c;               // sum of squares
  if (f == 4) { float m = a + b + c; return m*m*(1.0f/9.0f); } // m^2
  float d = a - c; return d*d;                       // (x0-x2)^2
}

// ---------------- kernel 1: fused separable 9x9 circular box + combine ----
__global__ __launch_bounds__(BLK1)
void k_box(const float* __restrict__ x,
           float* __restrict__ Apix,     // alpha-beta per pixel
           float* __restrict__ wpre,     // pre-exp weight per pixel
           double* __restrict__ blockW)  // per-block partial of sum(wpre)
{
  extern __shared__ float smem[];
  float* raw  = smem;                    // [c][hr][w]  CH*TROWS*WW
  float* fbuf = smem + CH*TROWS*WW;      // field values TROWS*WW
  float* hbuf = fbuf + TROWS*WW;         // horizontal sums TROWS*WW
  __shared__ float wred[BLK1/32];

  const int tid  = threadIdx.x;
  const int n    = blockIdx.x >> 5;      // /32 bands
  const int band = blockIdx.x & 31;
  const int r0   = band * BAND;

  // ---- cooperative load of raw tile (circular in h) ----
  const float* xn = x + (size_t)n * CH * HH * WW;
  for (int idx = tid; idx < CH*TROWS*WW; idx += BLK1) {
    int w  = idx & (WW - 1);
    int t  = idx >> 9;                   // c*TROWS + hr
    int hr = t % TROWS;
    int c  = t / TROWS;
    int gh = (r0 - RAD + hr) & (HH - 1);
    raw[idx] = xn[((size_t)c*HH + gh)*WW + w];
  }
  __syncthreads();

  const int lane  = tid & 31;
  const int wv    = tid >> 5;            // wave id 0..15
  const int mL    = lane & 15;
  const int khalf = (lane < 16) ? 0 : 2;

  float acc[6][2][8];                    // [field][col-tile][C/D vgpr]

#pragma unroll
  for (int f = 0; f < 6; ++f) {
    // field values for all halo rows (thread owns one column)
    for (int hr = 0; hr < TROWS; ++hr) {
      int i = hr*WW + tid;
      float a  = raw[i];
      float b  = raw[i +   TROWS*WW];
      float cc = raw[i + 2*TROWS*WW];
      fbuf[i] = fieldval(f, a, b, cc);
    }
    __syncthreads();

    // horizontal 9-tap circular sums
    for (int hr = 0; hr < TROWS; ++hr) {
      float s = 0.f;
#pragma unroll
      for (int dw = -RAD; dw <= RAD; ++dw)
        s += fbuf[hr*WW + ((tid + dw) & (WW - 1))];
      hbuf[hr*WW + tid] = s;
    }
    __syncthreads();

    // vertical 9-tap as WMMA: D(16x16) = Band(16x24) x Hsum(24x16), K chunks of 4
#pragma unroll
    for (int t2 = 0; t2 < 2; ++t2) {
      const int col = (wv*2 + t2)*16 + mL;
      v8f cacc = {0.f,0.f,0.f,0.f,0.f,0.f,0.f,0.f};
#if HAVE_WMMA_F32
#pragma unroll
      for (int kk = 0; kk < 6; ++kk) {
        const int k0 = kk*4 + khalf;     // this lane's K pair
        v2f A, B;
        // band matrix: A[m][k] = 1 iff m <= k <= m+8   (m = mL)
        A[0] = (k0     >= mL && k0     <= mL + 8) ? 1.0f : 0.0f;
        A[1] = (k0 + 1 >= mL && k0 + 1 <= mL + 8) ? 1.0f : 0.0f;
        B[0] = hbuf[(k0    )*WW + col];
        B[1] = hbuf[(k0 + 1)*WW + col];
        cacc = __builtin_amdgcn_wmma_f32_16x16x4_f32(
            false, A, false, B, (short)0, cacc, false, false);
      }
#else
#pragma unroll
      for (int j = 0; j < 8; ++j) {
        int m = j + ((lane < 16) ? 0 : 8);
        float s = 0.f;
        for (int k = m; k <= m + 8; ++k) s += hbuf[k*WW + col];
        cacc[j] = s;
      }
#endif
#pragma unroll
      for (int j = 0; j < 8; ++j) acc[f][t2][j] = cacc[j];
    }
    __syncthreads();   // hbuf/fbuf reused next field
  }

  // ---- per-pixel combine: D layout lane<16 -> rows j, lane>=16 -> rows j+8 ----
  const float inv81 = 1.0f / 81.0f;
  float wpart = 0.f;
#pragma unroll
  for (int t2 = 0; t2 < 2; ++t2) {
    const int col = (wv*2 + t2)*16 + mL;
#pragma unroll
    for (int j = 0; j < 8; ++j) {
      int row = j + ((lane < 16) ? 0 : 8);
      float b0  = acc[0][t2][j] * inv81;
      float b1  = acc[1][t2][j] * inv81;
      float b2  = acc[2][t2][j] * inv81;
      float bs  = acc[3][t2][j] * inv81;
      float bmm = acc[4][t2][j] * inv81;
      float bdd = acc[5][t2][j] * inv81;
      float alpha = (bs - (b0*b0 + b1*b1 + b2*b2)) * (1.0f/3.0f);
      float mmean = (b0 + b1 + b2) * (1.0f/3.0f);
      float beta  = bmm - mmean*mmean;
      float dmean = b0 - b2;
      float wp    = (2.0f/3.0f) * (bdd - dmean*dmean);
      size_t pix = ((size_t)n*HH + (r0 + row))*WW + col;
      Apix[pix] = alpha - beta;
      wpre[pix] = wp;
      wpart += wp;
    }
  }
  // block reduce wpart -> blockW (deterministic, no atomics)
  for (int off = 16; off; off >>= 1) wpart += __shfl_xor(wpart, off, 32);
  if (lane == 0) wred[wv] = wpart;
  __syncthreads();
  if (tid == 0) {
    double tw = 0.0;
    for (int i = 0; i < BLK1/32; ++i) tw += (double)wred[i];
    blockW[blockIdx.x] = tw;
  }
}

// ---------------- kernel 2: reduce 512 block partials -> Wsum ----
__global__ void k_redW(const double* __restrict__ blockW, double* __restrict__ Wsum) {
  __shared__ double s[512];
  int t = threadIdx.x;
  s[t] = blockW[t];
  __syncthreads();
  for (int off = 256; off; off >>= 1) {
    if (t < off) s[t] += s[t + off];
    __syncthreads();
  }
  if (t == 0) *Wsum = s[0];
}

// ---------------- kernel 3: exp weighting + partial sums ----
__global__ __launch_bounds__(BLK2)
void k_expsum(const float* __restrict__ Apix, const float* __restrict__ wpre,
              const double* __restrict__ Wsum,
              double* __restrict__ blockE, double* __restrict__ blockEA)
{
  const float coef = (float)(-2.0 * (double)(HH*WW) / *Wsum); // -GAMMA*w*h/W
  float se = 0.f, sea = 0.f;
  for (int i = blockIdx.x*BLK2 + threadIdx.x; i < NPIX; i += GRID2*BLK2) {
    float wp = wpre[i];
    float e  = __expf(coef * wp);
    se  += e;
    sea += e * Apix[i];
  }
  __shared__ float re[BLK2/32], ra[BLK2/32];
  for (int off = 16; off; off >>= 1) {
    se  += __shfl_xor(se,  off, 32);
    sea += __shfl_xor(sea, off, 32);
  }
  int lane = threadIdx.x & 31, wv = threadIdx.x >> 5;
  if (lane == 0) { re[wv] = se; ra[wv] = sea; }
  __syncthreads();
  if (threadIdx.x == 0) {
    double te = 0.0, ta = 0.0;
    for (int i = 0; i < BLK2/32; ++i) { te += (double)re[i]; ta += (double)ra[i]; }
    blockE[blockIdx.x]  = te;
    blockEA[blockIdx.x] = ta;
  }
}

// ---------------- kernel 4: final reduce + scalar result ----
__global__ void k_final(const double* __restrict__ blockE,
                        const double* __restrict__ blockEA,
                        float* __restrict__ out)
{
  __shared__ double se[1024], sa[1024];
  int t = threadIdx.x;
  se[t] = blockE[t]  + blockE[t + 1024];
  sa[t] = blockEA[t] + blockEA[t + 1024];
  __syncthreads();
  for (int off = 512; off; off >>= 1) {
    if (t < off) { se[t] += se[t + off]; sa[t] += sa[t + off]; }
    __syncthreads();
  }
  if (t == 0) out[0] = (float)(1.5 * sa[0] / se[0]);   // scale = 3/2 (REAL=False)
}

// ---------------- host launcher ----------------
extern "C" void kernel_launch(void* const* d_in, const int* in_sizes, int n_in,
                              void* d_out, int out_size, void* d_ws, size_t ws_size,
                              hipStream_t stream) {
  (void)in_sizes; (void)n_in; (void)out_size; (void)ws_size;
  const float* x = (const float*)d_in[0];
  float* out = (float*)d_out;
  char* ws = (char*)d_ws;

  // scratch layout (all partials fully rewritten every call -> deterministic)
  double* Wsum    = (double*)(ws + 0);
  double* blockW  = (double*)(ws + 1024);                    // 512 doubles
  double* blockE  = (double*)(ws + 8192);                    // 2048 doubles
  double* blockEA = (double*)(ws + 8192 + 16384);            // 2048 doubles
  float*  Apix    = (float*)(ws + 65536);                    // 16 MiB
  float*  wpreA   = (float*)(ws + 65536 + (size_t)NPIX*4);   // 16 MiB

  const size_t shmem = (size_t)(CH*TROWS + 2*TROWS) * WW * sizeof(float); // 240 KB

  k_box   <<<NB*(HH/BAND), BLK1, shmem, stream>>>(x, Apix, wpreA, blockW);
  k_redW  <<<1, 512, 0, stream>>>(blockW, Wsum);
  k_expsum<<<GRID2, BLK2, 0, stream>>>(Apix, wpreA, Wsum, blockE, blockEA);
  k_final <<<1, 1024, 0, stream>>>(blockE, blockEA, out);
}